// Ours_39977555591191
// MI455X (gfx1250) — compile-verified
//
#include <hip/hip_runtime.h>
#include <hip/hip_bf16.h>

typedef float v2f __attribute__((ext_vector_type(2)));
typedef float v8f __attribute__((ext_vector_type(8)));

// ---------------------------------------------------------------------------
// 1) score[i] = tanh(dot(x_i, w) / ||w||) via V_WMMA_F32_16X16X4_F32.
//    One wave handles 16 rows; B is w replicated across all 16 N-columns so
//    every column of D holds the dot product. A layout per ISA 7.12.2.
// ---------------------------------------------------------------------------
__global__ void kScore(const float* __restrict__ x, const float* __restrict__ w,
                       float* __restrict__ score, int N) {
    const int lane = threadIdx.x & 31;
    const int wib  = threadIdx.x >> 5;
    const long wave = (long)blockIdx.x * (blockDim.x >> 5) + wib;
    const long row0 = wave * 16;
    if (row0 >= N) return;                       // uniform per wave

    // ||w||^2 : 128 weights, 4 per lane, butterfly reduce
    float4 wv = ((const float4*)w)[lane];
    float ss = wv.x*wv.x + wv.y*wv.y + wv.z*wv.z + wv.w*wv.w;
    for (int o = 16; o > 0; o >>= 1) ss += __shfl_xor(ss, o, 32);
    const float invn = 1.0f / sqrtf(ss);

    const int m    = lane & 15;
    const int hsel = lane >> 4;                  // 0: K=k0,k0+1   1: K=k0+2,k0+3
    const float* xr = x + (row0 + m) * 128 + (hsel << 1);

    v8f acc = {0.f,0.f,0.f,0.f,0.f,0.f,0.f,0.f};
#pragma unroll
    for (int k0 = 0; k0 < 128; k0 += 4) {
        float2 av = *(const float2*)(xr + k0);
        v2f a; a.x = av.x; a.y = av.y;
        v2f b;
        b.x = w[k0 + (hsel << 1)];
        b.y = w[k0 + (hsel << 1) + 1];
        acc = __builtin_amdgcn_wmma_f32_16x16x4_f32(false, a, false, b,
                                                    (short)0, acc, false, false);
    }
    // C/D layout: VGPR r -> (M=r, lanes 0-15) / (M=r+8, lanes 16-31)
    float val  = acc[lane & 7];
    int   srcl = (lane < 8) ? lane : lane + 8;
    float dot  = __shfl(val, srcl, 32);
    if (lane < 16) score[row0 + lane] = tanhf(dot * invn);
}

// ---------------------------------------------------------------------------
// 2) Per-graph bitonic sort of 1024 (score desc, idx asc), emit perm/relabel.
// ---------------------------------------------------------------------------
__global__ void kTopK(const float* __restrict__ score, int* __restrict__ perm,
                      int* __restrict__ relabel, int K) {
    __shared__ float sk[1024];
    __shared__ int   si[1024];
    const int t = threadIdx.x;
    const int g = blockIdx.x;
    const long gbase = (long)g * 1024;
    sk[t] = score[gbase + t];
    si[t] = t;
    __syncthreads();
    for (int k = 2; k <= 1024; k <<= 1) {
        for (int j = k >> 1; j > 0; j >>= 1) {
            int ixj = t ^ j;
            if (ixj > t) {
                float ka = sk[t], kb = sk[ixj];
                int   ia = si[t], ib = si[ixj];
                bool aFirst  = (ka > kb) || (ka == kb && ia < ib); // descending order
                bool dirDesc = ((t & k) == 0);
                if (aFirst != dirDesc) { sk[t]=kb; sk[ixj]=ka; si[t]=ib; si[ixj]=ia; }
            }
            __syncthreads();
        }
    }
    if (t < K) perm[(long)g * K + t] = (int)(gbase + si[t]);
    relabel[gbase + si[t]] = (t < K) ? (g * K + t) : -1;
}

// ---------------------------------------------------------------------------
// 3) x_out[r] = x[perm[r]] * score[perm[r]]  (one row per wave, float4/lane)
// ---------------------------------------------------------------------------
__global__ void kGatherX(const float* __restrict__ x, const int* __restrict__ perm,
                         const float* __restrict__ score, float* __restrict__ out, int P) {
    const int lane = threadIdx.x & 31;
    const int wib  = threadIdx.x >> 5;
    const long row = (long)blockIdx.x * (blockDim.x >> 5) + wib;
    if (row >= P) return;
    const int   p = perm[row];
    const float s = score[p];
    float4 v = ((const float4*)(x + (long)p * 128))[lane];
    v.x *= s; v.y *= s; v.z *= s; v.w *= s;
    ((float4*)(out + row * 128))[lane] = v;
}

// ---------------------------------------------------------------------------
// 4) Kept-edge count per 4096-edge chunk
// ---------------------------------------------------------------------------
__global__ void kEdgeCount(const int* __restrict__ esrc, const int* __restrict__ edst,
                           const int* __restrict__ relabel, unsigned* __restrict__ bcount,
                           long E) {
    __shared__ unsigned red[256];
    unsigned cnt = 0;
    const long base = (long)blockIdx.x * 4096;
    for (int r = 0; r < 16; ++r) {
        long e = base + r * 256 + threadIdx.x;
        if (e < E)
            cnt += (relabel[esrc[e]] >= 0) && (relabel[edst[e]] >= 0);
    }
    red[threadIdx.x] = cnt;
    __syncthreads();
    for (int o = 128; o > 0; o >>= 1) {
        if (threadIdx.x < (unsigned)o) red[threadIdx.x] += red[threadIdx.x + o];
        __syncthreads();
    }
    if (threadIdx.x == 0) bcount[blockIdx.x] = red[0];
}

// 5) scan chunk counts (nb <= 1024), emit M and topk = floor(0.9*M)
__global__ void kScanC(const unsigned* __restrict__ bcount, unsigned* __restrict__ boff,
                       int* __restrict__ counters, int nb) {
    __shared__ unsigned s[1024];
    const int t = threadIdx.x;
    unsigned c = (t < nb) ? bcount[t] : 0u;
    s[t] = c; __syncthreads();
    for (int o = 1; o < 1024; o <<= 1) {
        unsigned v = (t >= o) ? s[t - o] : 0u;
        __syncthreads();
        s[t] += v;
        __syncthreads();
    }
    if (t < nb) boff[t] = s[t] - c;
    if (t == 1023) {
        int M = (int)s[1023];
        counters[0] = M;
        counters[1] = (int)(0.9 * (double)M);
    }
}

// ---------------------------------------------------------------------------
// 6) Order-preserving compaction of (key=~bits(|ds|), origEdgeIdx)
// ---------------------------------------------------------------------------
__global__ void kEdgeScatter(const int* __restrict__ esrc, const int* __restrict__ edst,
                             const int* __restrict__ relabel, const float* __restrict__ score,
                             const unsigned* __restrict__ boff,
                             unsigned* __restrict__ keyA, unsigned* __restrict__ idxA, long E) {
    __shared__ unsigned wcnt[8];
    __shared__ unsigned running;
    const int t = threadIdx.x, lane = t & 31, w = t >> 5;
    if (t == 0) running = 0;
    const long base = (long)blockIdx.x * 4096;
    const unsigned gbase = boff[blockIdx.x];
    __syncthreads();
    for (int r = 0; r < 16; ++r) {
        long e = base + r * 256 + t;
        bool keep = false; float sc = 0.f;
        if (e < E) {
            int a = esrc[e], b = edst[e];
            if (relabel[a] >= 0 && relabel[b] >= 0) {
                keep = true;
                sc = fabsf(score[a] - score[b]);
            }
        }
        unsigned m    = (unsigned)__ballot(keep);
        unsigned rank = __popc(m & ((1u << lane) - 1u));
        if (lane == 0) wcnt[w] = __popc(m);
        __syncthreads();
        unsigned pre = 0;
        for (int i = 0; i < w; ++i) pre += wcnt[i];
        if (keep) {
            unsigned pos = gbase + running + pre + rank;
            keyA[pos] = ~__float_as_uint(sc);   // ascending key == descending score
            idxA[pos] = (unsigned)e;
        }
        __syncthreads();
        if (t == 0) { unsigned tot = 0; for (int i = 0; i < 8; ++i) tot += wcnt[i]; running += tot; }
        __syncthreads();
    }
}

// ---------------------------------------------------------------------------
// 7) Stable LSD radix sort, 8-bit digits
// ---------------------------------------------------------------------------
__global__ void kHist(const unsigned* __restrict__ key, const int* __restrict__ counters,
                      unsigned* __restrict__ hist, int shift) {
    __shared__ unsigned h[256];
    h[threadIdx.x] = 0; __syncthreads();
    const int M = counters[0];
    const long base = (long)blockIdx.x * 4096;
    for (int r = 0; r < 16; ++r) {
        long i = base + r * 256 + threadIdx.x;
        if (i < M) atomicAdd(&h[(key[i] >> shift) & 255u], 1u);
    }
    __syncthreads();
    hist[(long)blockIdx.x * 256 + threadIdx.x] = h[threadIdx.x];
}

__global__ void kScanDigits(const unsigned* __restrict__ hist, unsigned* __restrict__ baseArr,
                            int nb) {
    __shared__ unsigned s[256];
    const int d = threadIdx.x;
    unsigned tot = 0;
    for (int b = 0; b < nb; ++b) tot += hist[(long)b * 256 + d];
    s[d] = tot; __syncthreads();
    unsigned c = tot;
    for (int o = 1; o < 256; o <<= 1) {
        unsigned v = (d >= o) ? s[d - o] : 0u;
        __syncthreads();
        s[d] += v;
        __syncthreads();
    }
    unsigned run = s[d] - c;                    // exclusive base of digit d
    for (int b = 0; b < nb; ++b) {
        baseArr[(long)b * 256 + d] = run;
        run += hist[(long)b * 256 + d];
    }
}

__global__ void kScatter(const unsigned* __restrict__ keyIn, const unsigned* __restrict__ idxIn,
                         unsigned* __restrict__ keyOut, unsigned* __restrict__ idxOut,
                         const unsigned* __restrict__ baseArr, const int* __restrict__ counters,
                         int shift) {
    __shared__ unsigned cursor[256];
    const int t = threadIdx.x, lane = t & 31, w = t >> 5;
    cursor[t] = baseArr[(long)blockIdx.x * 256 + t];
    __syncthreads();
    const int M = counters[0];
    const long cbase = (long)blockIdx.x * 4096;
    for (int r = 0; r < 16; ++r) {
        long i = cbase + r * 256 + t;
        bool valid = (i < M);
        unsigned key = valid ? keyIn[i] : 0u;
        unsigned pay = valid ? idxIn[i] : 0u;
        unsigned d   = (key >> shift) & 255u;
        // in-wave same-digit match via 8 ballots
        unsigned m = 0xffffffffu;
#pragma unroll
        for (int k = 0; k < 8; ++k) {
            unsigned bit = (d >> k) & 1u;
            unsigned bm  = (unsigned)__ballot(bit != 0);
            m &= bit ? bm : ~bm;
        }
        m &= (unsigned)__ballot(valid);
        unsigned rank = __popc(m & ((1u << lane) - 1u));
        unsigned cnt  = __popc(m);
        // wave-serialized stable cursor claim
        for (int wq = 0; wq < 8; ++wq) {
            if (w == wq && valid) {
                unsigned bp  = cursor[d];
                unsigned pos = bp + rank;
                if (rank == cnt - 1) cursor[d] = bp + cnt;
                keyOut[pos] = key;
                idxOut[pos] = pay;
            }
            __syncthreads();
        }
    }
}

// ---------------------------------------------------------------------------
// 8) Emit edge_index_out (relabelled, topk edges) and batch_out
// ---------------------------------------------------------------------------
__global__ void kFinal(const unsigned* __restrict__ idxA, const int* __restrict__ relabel,
                       const int* __restrict__ esrc, const int* __restrict__ edst,
                       const int* __restrict__ counters, float* __restrict__ out,
                       long P, int K) {
    const long j = (long)blockIdx.x * blockDim.x + threadIdx.x;
    const int topk = counters[1];
    const long base = P * 128;
    if (j < topk) {
        unsigned e = idxA[j];
        out[base + j]        = (float)relabel[esrc[e]];
        out[base + topk + j] = (float)relabel[edst[e]];
    }
    if (j < P) out[base + 2L * topk + j] = (float)(j / K);
}

extern "C" void kernel_launch(void* const* d_in, const int* in_sizes, int n_in,
                              void* d_out, int out_size, void* d_ws, size_t ws_size,
                              hipStream_t stream) {
    const float* x = (const float*)d_in[0];
    const float* w = (const float*)d_in[1];
    const int*  ei = (const int*)d_in[2];

    const int  Dd  = 128;
    const long N   = in_sizes[0] / Dd;          // 262144
    const long E   = in_sizes[2] / 2;           // 4,000,000
    const int  NPG = 1024;
    const long B   = N / NPG;                   // 256
    const int  K   = (NPG + 1) / 2;             // ceil(0.5*NPG) = 512
    const long P   = B * K;                     // 131072

    const int* esrc = ei;
    const int* edst = ei + E;
    float* out = (float*)d_out;

    // workspace carve (u32 units)
    unsigned* Wp = (unsigned*)d_ws;
    size_t off = 0;
    float* score   = (float*)(Wp + off); off += (size_t)N;
    int*   relabel = (int*)(Wp + off);   off += (size_t)N;
    int*   perm    = (int*)(Wp + off);   off += (size_t)P;
    const int nbC  = (int)((E + 4095) / 4096);
    unsigned* bcount  = Wp + off; off += (size_t)nbC;
    unsigned* boff    = Wp + off; off += (size_t)nbC;
    int*      counters= (int*)(Wp + off); off += 8;
    unsigned* hist    = Wp + off; off += (size_t)nbC * 256;
    unsigned* baseArr = Wp + off; off += (size_t)nbC * 256;
    unsigned* keyA = Wp + off; off += (size_t)E;
    unsigned* idxA = Wp + off; off += (size_t)E;
    unsigned* keyB = Wp + off; off += (size_t)E;
    unsigned* idxB = Wp + off; off += (size_t)E;

    kScore   <<<(int)(N / 128), 256, 0, stream>>>(x, w, score, (int)N);
    kTopK    <<<(int)B, 1024, 0, stream>>>(score, perm, relabel, K);
    kGatherX <<<(int)((P + 7) / 8), 256, 0, stream>>>(x, perm, score, out, (int)P);

    kEdgeCount  <<<nbC, 256, 0, stream>>>(esrc, edst, relabel, bcount, E);
    kScanC      <<<1, 1024, 0, stream>>>(bcount, boff, counters, nbC);
    kEdgeScatter<<<nbC, 256, 0, stream>>>(esrc, edst, relabel, score, boff, keyA, idxA, E);

    // stable 32-bit LSD radix sort: A->B->A->B->A
    kHist      <<<nbC, 256, 0, stream>>>(keyA, counters, hist, 0);
    kScanDigits<<<1, 256, 0, stream>>>(hist, baseArr, nbC);
    kScatter   <<<nbC, 256, 0, stream>>>(keyA, idxA, keyB, idxB, baseArr, counters, 0);

    kHist      <<<nbC, 256, 0, stream>>>(keyB, counters, hist, 8);
    kScanDigits<<<1, 256, 0, stream>>>(hist, baseArr, nbC);
    kScatter   <<<nbC, 256, 0, stream>>>(keyB, idxB, keyA, idxA, baseArr, counters, 8);

    kHist      <<<nbC, 256, 0, stream>>>(keyA, counters, hist, 16);
    kScanDigits<<<1, 256, 0, stream>>>(hist, baseArr, nbC);
    kScatter   <<<nbC, 256, 0, stream>>>(keyA, idxA, keyB, idxB, baseArr, counters, 16);

    kHist      <<<nbC, 256, 0, stream>>>(keyB, counters, hist, 24);
    kScanDigits<<<1, 256, 0, stream>>>(hist, baseArr, nbC);
    kScatter   <<<nbC, 256, 0, stream>>>(keyB, idxB, keyA, idxA, baseArr, counters, 24);

    kFinal<<<(int)((E + 255) / 256), 256, 0, stream>>>(idxA, relabel, esrc, edst,
                                                       counters, out, P, K);
}